// GCN_1425929142718
// MI455X (gfx1250) — compile-verified
//
#include <hip/hip_runtime.h>

typedef __attribute__((ext_vector_type(2))) float v2f;
typedef __attribute__((ext_vector_type(8))) float v8f;

#define IN_DIM 128
#define HID    128
#define OUTD   64

// ---------------------------------------------------------------------------
// init: deg = 1.0 (self loop contribution), agg = 0
// ---------------------------------------------------------------------------
__global__ void gcn_init_ws(float* __restrict__ agg, float* __restrict__ deg,
                            int n_nodes) {
  int idx = blockIdx.x * blockDim.x + threadIdx.x;
  int total = n_nodes * HID;
  if (idx < total) agg[idx] = 0.0f;
  if (idx < n_nodes) deg[idx] = 1.0f;
}

// ---------------------------------------------------------------------------
// degree count over dst (edge_index is int64 in the reference)
// ---------------------------------------------------------------------------
__global__ void gcn_degree(const long long* __restrict__ dst,
                           float* __restrict__ deg, int n_edges) {
  int e = blockIdx.x * blockDim.x + threadIdx.x;
  if (e < n_edges) atomicAdd(&deg[(int)dst[e]], 1.0f);
}

// ---------------------------------------------------------------------------
// dinv = rsqrt(deg) in place (deg >= 1 always due to self loops)
// ---------------------------------------------------------------------------
__global__ void gcn_rsqrt(float* __restrict__ deg, int n_nodes) {
  int i = blockIdx.x * blockDim.x + threadIdx.x;
  if (i < n_nodes) deg[i] = rsqrtf(deg[i]);
}

// ---------------------------------------------------------------------------
// fp32 WMMA GEMM: C[M,N] = A[M,128] * B[128,N] (+ bias), N compile-time.
// One block = one 32-row panel (two 16-row tiles per wave, sharing the B
// fragment). Wave w owns columns [16w, 16w+16). blockDim.x = 32 * (N/16).
// All immediate-offset loads: kp*N folds into the 24-bit IOFFSET.
// M handled here must be a multiple of 32 (tail via gcn_gemm_tail).
// ---------------------------------------------------------------------------
template <int N>
__global__ void gcn_gemm_wmma(const float* __restrict__ A,
                              const float* __restrict__ B,
                              const float* __restrict__ bias,
                              float* __restrict__ C) {
  const int lane = threadIdx.x & 31;
  const int wave = threadIdx.x >> 5;
  const int m0   = blockIdx.x << 5;   // 32 rows per block
  const int n0   = wave << 4;
  const int l16  = lane & 15;         // M index (A frag) / N index (B frag)
  const int hi   = lane >> 4;         // 0: K pair {0,1}; 1: K pair {2,3}
  const int kh   = hi << 1;

  const float* __restrict__ A0 = A + (size_t)(m0 + l16) * IN_DIM + kh;
  const float* __restrict__ A1 = A0 + 16 * IN_DIM;
  const float* __restrict__ Bc = B + (size_t)kh * N + (n0 + l16);

  v8f acc0 = {};
  v8f acc1 = {};
#pragma unroll
  for (int k0 = 0; k0 < IN_DIM; k0 += 4) {
    v2f a0 = *(const v2f*)(A0 + k0);          // 8B-aligned (kh even, k0 % 4 == 0)
    v2f a1 = *(const v2f*)(A1 + k0);
    v2f b;
    b.x = Bc[k0 * N];                         // immediate offsets (N constexpr)
    b.y = Bc[k0 * N + N];
    acc0 = __builtin_amdgcn_wmma_f32_16x16x4_f32(
        false, a0, false, b, (short)0, acc0, false, false);
    acc1 = __builtin_amdgcn_wmma_f32_16x16x4_f32(
        false, a1, false, b, (short)0, acc1, false, false);
  }

  const int col = n0 + l16;
  const float bv = bias ? bias[col] : 0.0f;
  float* __restrict__ C0 = C + (size_t)(m0 + (hi << 3)) * N + col;
  float* __restrict__ C1 = C0 + (size_t)16 * N;
#pragma unroll
  for (int r = 0; r < 8; ++r) {
    C0[r * N] = acc0[r] + bv;                 // immediate offsets
    C1[r * N] = acc1[r] + bv;
  }
}

// Scalar fallback for M-remainder rows (not launched when M % 32 == 0).
__global__ void gcn_gemm_tail(const float* __restrict__ A,
                              const float* __restrict__ B,
                              const float* __restrict__ bias,
                              float* __restrict__ C,
                              int row_start, int rows, int N) {
  int idx = blockIdx.x * blockDim.x + threadIdx.x;
  if (idx >= rows * N) return;
  int r = row_start + idx / N;
  int c = idx % N;
  float s = bias ? bias[c] : 0.0f;
  for (int k = 0; k < IN_DIM; ++k)
    s += A[(size_t)r * IN_DIM + k] * B[(size_t)k * N + c];
  C[(size_t)r * N + c] = s;
}

// ---------------------------------------------------------------------------
// edge scatter: agg[dst] += h[src] * dinv[src] * dinv[dst]
// 32 lanes per edge, 4 consecutive features per lane (float4 gather of h).
// ---------------------------------------------------------------------------
__global__ void gcn_scatter(const long long* __restrict__ src,
                            const long long* __restrict__ dst,
                            const float* __restrict__ dinv,
                            const float* __restrict__ h,
                            float* __restrict__ agg, int n_edges) {
  long long tid = (long long)blockIdx.x * blockDim.x + threadIdx.x;
  int e = (int)(tid >> 5);
  int q = ((int)tid & 31) << 2;               // feature base: 0..124
  if (e >= n_edges) return;
  int s = (int)src[e];
  int d = (int)dst[e];
  float nrm = dinv[s] * dinv[d];
  const float4 hv = *(const float4*)(h + (size_t)s * HID + q);
  float* __restrict__ out = agg + (size_t)d * HID + q;
  atomicAdd(out + 0, hv.x * nrm);
  atomicAdd(out + 1, hv.y * nrm);
  atomicAdd(out + 2, hv.z * nrm);
  atomicAdd(out + 3, hv.w * nrm);
}

// ---------------------------------------------------------------------------
// finalize layer 1: agg = relu(agg + h * (1/deg) + b1)
// (self-loop norm = dinv[i]^2 = 1/deg[i]); done in place in agg.
// ---------------------------------------------------------------------------
__global__ void gcn_selfloop_bias_relu(float* __restrict__ agg,
                                       const float* __restrict__ h,
                                       const float* __restrict__ dinv,
                                       const float* __restrict__ b1,
                                       int n_nodes) {
  int idx = blockIdx.x * blockDim.x + threadIdx.x;
  if (idx >= n_nodes * HID) return;
  int i = idx >> 7;                            // HID == 128
  int c = idx & 127;
  float di = dinv[i];
  float v = agg[idx] + h[idx] * di * di + b1[c];
  agg[idx] = v > 0.0f ? v : 0.0f;
}

// ---------------------------------------------------------------------------
extern "C" void kernel_launch(void* const* d_in, const int* in_sizes, int n_in,
                              void* d_out, int out_size, void* d_ws,
                              size_t ws_size, hipStream_t stream) {
  const float*     x  = (const float*)d_in[0];
  const float*     W1 = (const float*)d_in[1];
  const float*     b1 = (const float*)d_in[2];
  const float*     W2 = (const float*)d_in[3];
  const float*     b2 = (const float*)d_in[4];
  const long long* ei = (const long long*)d_in[5];   // (2, E) int64

  const int n_nodes = in_sizes[0] / IN_DIM;
  const int n_edges = in_sizes[5] / 2;
  const long long* src = ei;
  const long long* dst = ei + n_edges;

  // workspace layout: deg | h | agg
  float* deg = (float*)d_ws;
  float* h   = deg + n_nodes;
  float* agg = h + (size_t)n_nodes * HID;
  float* out = (float*)d_out;

  const int totalNH = n_nodes * HID;

  // 1) init workspace (deg starts at 1.0 for the self loop)
  gcn_init_ws<<<(totalNH + 255) / 256, 256, 0, stream>>>(agg, deg, n_nodes);

  // 2) degree count over dst
  gcn_degree<<<(n_edges + 255) / 256, 256, 0, stream>>>(dst, deg, n_edges);

  // 3) dinv = rsqrt(deg) in place
  gcn_rsqrt<<<(n_nodes + 255) / 256, 256, 0, stream>>>(deg, n_nodes);

  // 4) h = x @ W1   (fp32 WMMA; 8 waves = 8 col tiles over N=128, 32 rows/blk)
  const int mblocks = n_nodes / 32;
  if (mblocks > 0)
    gcn_gemm_wmma<HID><<<mblocks, 8 * 32, 0, stream>>>(x, W1, nullptr, h);
  const int mrem = n_nodes - mblocks * 32;
  if (mrem > 0)
    gcn_gemm_tail<<<(mrem * HID + 255) / 256, 256, 0, stream>>>(
        x, W1, nullptr, h, mblocks * 32, mrem, HID);

  // 5) edge scatter-add into agg
  {
    long long threads = (long long)n_edges * 32;
    int blocks = (int)((threads + 255) / 256);
    gcn_scatter<<<blocks, 256, 0, stream>>>(src, dst, deg, h, agg, n_edges);
  }

  // 6) agg = relu(agg + h/deg + b1)
  gcn_selfloop_bias_relu<<<(totalNH + 255) / 256, 256, 0, stream>>>(
      agg, h, deg, b1, n_nodes);

  // 7) out = agg @ W2 + b2  (fp32 WMMA; 4 waves over N=64, 32 rows/blk)
  if (mblocks > 0)
    gcn_gemm_wmma<OUTD><<<mblocks, 4 * 32, 0, stream>>>(agg, W2, b2, out);
  if (mrem > 0)
    gcn_gemm_tail<<<(mrem * OUTD + 255) / 256, 256, 0, stream>>>(
        agg, W2, b2, out, mblocks * 32, mrem, OUTD);
}